// Self_Attn_77756087927687
// MI455X (gfx1250) — compile-verified
//
#include <hip/hip_runtime.h>

typedef __attribute__((ext_vector_type(2))) float v2f;
typedef __attribute__((ext_vector_type(8))) float v8f;

#define NPIX 4096   // H*W
#define CCH  64     // C
#define ECH  8      // E = C/8
#define BB   4      // batch

// ---------------------------------------------------------------------------
// Kernel 1: fused 1x1-conv projections  f/g/h = W{k,q,v} @ x + b{k,q,v}
// One thread per (b, n). Weights (3*512 f32) staged in LDS.
// ---------------------------------------------------------------------------
__global__ __launch_bounds__(256) void proj_kernel(
    const float* __restrict__ x,
    const float* __restrict__ Wk, const float* __restrict__ bk,
    const float* __restrict__ Wq, const float* __restrict__ bq,
    const float* __restrict__ Wv, const float* __restrict__ bv,
    float* __restrict__ f, float* __restrict__ g, float* __restrict__ h)
{
    __shared__ float sWk[ECH * CCH], sWq[ECH * CCH], sWv[ECH * CCH];
    __shared__ float sbk[ECH], sbq[ECH], sbv[ECH];
    int tid = threadIdx.x;
    for (int t = tid; t < ECH * CCH; t += 256) {
        sWk[t] = Wk[t]; sWq[t] = Wq[t]; sWv[t] = Wv[t];
    }
    if (tid < ECH) { sbk[tid] = bk[tid]; sbq[tid] = bq[tid]; sbv[tid] = bv[tid]; }
    __syncthreads();

    int idx = blockIdx.x * 256 + tid;           // b*NPIX + n
    int b = idx >> 12, n = idx & (NPIX - 1);
    const float* xp = x + ((size_t)b * CCH) * NPIX + n;

    float af[ECH], ag[ECH], ah[ECH];
#pragma unroll
    for (int e = 0; e < ECH; ++e) { af[e] = sbk[e]; ag[e] = sbq[e]; ah[e] = sbv[e]; }

    for (int c = 0; c < CCH; ++c) {
        float xv = xp[(size_t)c * NPIX];
#pragma unroll
        for (int e = 0; e < ECH; ++e) {
            af[e] = fmaf(sWk[e * CCH + c], xv, af[e]);
            ag[e] = fmaf(sWq[e * CCH + c], xv, ag[e]);
            ah[e] = fmaf(sWv[e * CCH + c], xv, ah[e]);
        }
    }
    size_t base = ((size_t)b * ECH) * NPIX + n;
#pragma unroll
    for (int e = 0; e < ECH; ++e) {
        f[base + (size_t)e * NPIX] = af[e];
        g[base + (size_t)e * NPIX] = ag[e];
        h[base + (size_t)e * NPIX] = ah[e];
    }
}

// ---------------------------------------------------------------------------
// Kernel 2: s = f^T g  via V_WMMA_F32_16X16X4_F32 (two K=4 steps cover K=8).
// One wave per 16(i) x 256(j) strip; D written directly into beta output.
// A layout (f32 16x4): lane m=lane&15, kh=lane>>4; VGPR0=k(2kh), VGPR1=k(2kh+1).
// B layout symmetric over n. D: row = r + 8*kh, col = lane&15.
// ---------------------------------------------------------------------------
__global__ __launch_bounds__(256) void sgemm_kernel(
    const float* __restrict__ f, const float* __restrict__ g,
    float* __restrict__ S)
{
    int wid  = blockIdx.x * 8 + (threadIdx.x >> 5);  // 16384 waves total
    int lane = threadIdx.x & 31;
    int jStrip = wid & 15;
    int iTile  = (wid >> 4) & 255;
    int b      = wid >> 12;
    int i0  = iTile << 4;
    int j0b = jStrip << 8;
    int m  = lane & 15;
    int kh = lane >> 4;

    const float* F = f + (size_t)b * ECH * NPIX;
    const float* G = g + (size_t)b * ECH * NPIX;
    float* Sp = S + (size_t)b * NPIX * NPIX;

    v2f aLo, aHi;
    aLo.x = F[(2 * kh + 0) * NPIX + i0 + m];
    aLo.y = F[(2 * kh + 1) * NPIX + i0 + m];
    aHi.x = F[(2 * kh + 4) * NPIX + i0 + m];
    aHi.y = F[(2 * kh + 5) * NPIX + i0 + m];

    for (int jt = 0; jt < 16; ++jt) {
        int j0 = j0b + (jt << 4);
        v2f bLo, bHi;
        bLo.x = G[(2 * kh + 0) * NPIX + j0 + m];
        bLo.y = G[(2 * kh + 1) * NPIX + j0 + m];
        bHi.x = G[(2 * kh + 4) * NPIX + j0 + m];
        bHi.y = G[(2 * kh + 5) * NPIX + j0 + m];

        v8f c = {};
        c = __builtin_amdgcn_wmma_f32_16x16x4_f32(false, aLo, false, bLo,
                                                  (short)0, c, false, false);
        c = __builtin_amdgcn_wmma_f32_16x16x4_f32(false, aHi, false, bHi,
                                                  (short)0, c, false, false);
#pragma unroll
        for (int r = 0; r < 8; ++r)
            Sp[(size_t)(i0 + r + 8 * kh) * NPIX + (j0 + m)] = c[r];
    }
}

// ---------------------------------------------------------------------------
// Kernel 3: column softmax over i (in place on beta) fused with
//           v[b,e,j] = sum_i h[b,e,i] * beta[b,i,j].
// One thread per (b, j): consecutive threads hit consecutive addresses.
// ---------------------------------------------------------------------------
__global__ __launch_bounds__(256) void softmax_v_kernel(
    float* __restrict__ S, const float* __restrict__ h, float* __restrict__ v)
{
    int idx = blockIdx.x * 256 + threadIdx.x;    // b*NPIX + j
    int b = idx >> 12, j = idx & (NPIX - 1);
    float* Sp = S + (size_t)b * NPIX * NPIX + j;

    // pass 1: online max + sum(exp)
    float mx = -__builtin_inff();
    float l  = 0.0f;
    for (int i = 0; i < NPIX; ++i) {
        float s  = Sp[(size_t)i * NPIX];
        float mn = fmaxf(mx, s);
        l  = l * __expf(mx - mn) + __expf(s - mn);
        mx = mn;
    }
    float inv = 1.0f / l;

    // pass 2: write beta, accumulate v
    const float* H = h + (size_t)b * ECH * NPIX;
    float acc[ECH] = {0.f, 0.f, 0.f, 0.f, 0.f, 0.f, 0.f, 0.f};
    for (int i = 0; i < NPIX; ++i) {
        float s  = Sp[(size_t)i * NPIX];
        float be = __expf(s - mx) * inv;
        Sp[(size_t)i * NPIX] = be;
#pragma unroll
        for (int e = 0; e < ECH; ++e)
            acc[e] = fmaf(H[(size_t)e * NPIX + i], be, acc[e]);
    }
#pragma unroll
    for (int e = 0; e < ECH; ++e)
        v[((size_t)b * ECH + e) * NPIX + j] = acc[e];
}

// ---------------------------------------------------------------------------
// Kernel 4: o = Wo v + bo ; y = leaky_relu(gamma*o + x, 0.01)
// One thread per (b, c, n) == flat x/y index.
// ---------------------------------------------------------------------------
__global__ __launch_bounds__(256) void out_kernel(
    const float* __restrict__ v, const float* __restrict__ Wo,
    const float* __restrict__ bo, const float* __restrict__ gamma,
    const float* __restrict__ x, float* __restrict__ y)
{
    int idx = blockIdx.x * 256 + threadIdx.x;
    int n = idx & (NPIX - 1);
    int c = (idx >> 12) & (CCH - 1);
    int b = idx >> 18;

    float acc = bo[c];
    const float* vp = v + (size_t)b * ECH * NPIX + n;
#pragma unroll
    for (int e = 0; e < ECH; ++e)
        acc = fmaf(Wo[c * ECH + e], vp[(size_t)e * NPIX], acc);

    float yy = fmaf(gamma[0], acc, x[idx]);
    y[idx] = (yy >= 0.0f) ? yy : 0.01f * yy;
}

// ---------------------------------------------------------------------------
extern "C" void kernel_launch(void* const* d_in, const int* in_sizes, int n_in,
                              void* d_out, int out_size, void* d_ws, size_t ws_size,
                              hipStream_t stream)
{
    const float* x     = (const float*)d_in[0];
    const float* Wk    = (const float*)d_in[1];
    const float* bk    = (const float*)d_in[2];
    const float* Wq    = (const float*)d_in[3];
    const float* bq    = (const float*)d_in[4];
    const float* Wv    = (const float*)d_in[5];
    const float* bv    = (const float*)d_in[6];
    const float* Wo    = (const float*)d_in[7];
    const float* bo    = (const float*)d_in[8];
    const float* gamma = (const float*)d_in[9];

    float* y    = (float*)d_out;                        // [4,64,64,64]
    float* beta = y + (size_t)BB * CCH * NPIX;          // [4,4096,4096]

    float* f = (float*)d_ws;                            // [4,8,4096] each
    float* g = f + (size_t)BB * ECH * NPIX;
    float* h = g + (size_t)BB * ECH * NPIX;
    float* v = h + (size_t)BB * ECH * NPIX;

    proj_kernel<<<(BB * NPIX) / 256, 256, 0, stream>>>(x, Wk, bk, Wq, bq, Wv, bv, f, g, h);
    sgemm_kernel<<<(BB * 256 * 16) / 8, 256, 0, stream>>>(f, g, beta);
    softmax_v_kernel<<<(BB * NPIX) / 256, 256, 0, stream>>>(beta, h, v);
    out_kernel<<<(BB * CCH * NPIX) / 256, 256, 0, stream>>>(v, Wo, bo, gamma, x, y);
}